// InductionModule_19043884991115
// MI455X (gfx1250) — compile-verified
//
#include <hip/hip_runtime.h>
#include <math.h>

#define CLS 64
#define KSH 512
#define DIM 2048

typedef __attribute__((ext_vector_type(2))) float v2f;
typedef __attribute__((ext_vector_type(8))) float v8f;

// ---------------------------------------------------------------------------
// Kernel 1: per-row squash scale s[c,k] = ||z|| / (1 + ||z||^2), and b = 0.
// One wave32 per row of 2048 floats; 8 waves per block.
// ---------------------------------------------------------------------------
__global__ __launch_bounds__(256) void scales_kernel(const float* __restrict__ z,
                                                     float* __restrict__ s,
                                                     float* __restrict__ b) {
  const int wave = threadIdx.x >> 5;
  const int lane = threadIdx.x & 31;
  const int row  = blockIdx.x * 8 + wave;
  const float* zr = z + (size_t)row * DIM;

  float acc = 0.0f;
#pragma unroll 8
  for (int j = lane; j < DIM; j += 32) {
    float v = zr[j];
    acc = fmaf(v, v, acc);
  }
  // wave32 butterfly reduce
  for (int off = 16; off > 0; off >>= 1)
    acc += __shfl_xor(acc, off, 32);

  if (lane == 0) {
    float n2 = acc;
    s[row] = sqrtf(n2) / (1.0f + n2);
    b[row] = 0.0f;
  }
}

// ---------------------------------------------------------------------------
// Kernel 2: per class: w = softmax(b) * s ; c = squash(sum_k w_k * z_k).
// One 1024-thread block per class; each thread owns 2 d-columns.
// Pure streaming pass -> VALU FMA + global prefetch 16 rows ahead.
// ---------------------------------------------------------------------------
__global__ __launch_bounds__(1024) void cvec_kernel(const float* __restrict__ z,
                                                    const float* __restrict__ s,
                                                    const float* __restrict__ b,
                                                    float* __restrict__ cout) {
  __shared__ float w[KSH];
  __shared__ float red[1024];

  const int cls = blockIdx.x;
  const int tid = threadIdx.x;
  const float* zc = z + (size_t)cls * KSH * DIM;

  // ---- softmax over K=512 logits, folded with per-row scale s ----
  float bv = (tid < KSH) ? b[cls * KSH + tid] : -1e30f;
  red[tid] = bv;
  __syncthreads();
  for (int st = 512; st > 0; st >>= 1) {
    if (tid < st) red[tid] = fmaxf(red[tid], red[tid + st]);
    __syncthreads();
  }
  const float bmax = red[0];
  __syncthreads();

  float e = (tid < KSH) ? __expf(bv - bmax) : 0.0f;
  red[tid] = e;
  __syncthreads();
  for (int st = 512; st > 0; st >>= 1) {
    if (tid < st) red[tid] += red[tid + st];
    __syncthreads();
  }
  const float esum = red[0];
  __syncthreads();
  if (tid < KSH) w[tid] = (e / esum) * s[cls * KSH + tid];
  __syncthreads();

  // ---- c_d = sum_k w_k * z[c,k,d]  (coalesced stream over 4 MB slab) ----
  const int d0 = tid;
  const int d1 = tid + 1024;
  float a0 = 0.0f, a1 = 0.0f;
  for (int k = 0; k < KSH; ++k) {
    const float* zr = zc + (size_t)k * DIM;
    if (k + 16 < KSH)
      __builtin_prefetch(zr + (size_t)16 * DIM + d0, 0, 0);  // global_prefetch_b8
    const float wv = w[k];
    a0 = fmaf(wv, zr[d0], a0);
    a1 = fmaf(wv, zr[d1], a1);
  }

  // ---- squash over full D across the block ----
  red[tid] = fmaf(a0, a0, a1 * a1);
  __syncthreads();
  for (int st = 512; st > 0; st >>= 1) {
    if (tid < st) red[tid] += red[tid + st];
    __syncthreads();
  }
  const float n2 = red[0];
  const float scale = sqrtf(n2) / (1.0f + n2);
  cout[cls * DIM + d0] = a0 * scale;
  cout[cls * DIM + d1] = a1 * scale;
}

// ---------------------------------------------------------------------------
// Kernel 3: b[c,k] += s[c,k] * (z[c,k,:] . c[c,:])  via V_WMMA_F32_16X16X4_F32.
// A = 16 k-rows x 4 d-cols of Z; B = c-segment replicated into all 16 columns
// (every output column = the 16 partial dots; harvest column 0 from lanes 0/16).
// 8 waves per block, 16 rows per wave, 4 blocks per class; c staged in LDS.
// ---------------------------------------------------------------------------
__global__ __launch_bounds__(256) void bupd_kernel(const float* __restrict__ z,
                                                   const float* __restrict__ s,
                                                   const float* __restrict__ cvec,
                                                   float* __restrict__ b) {
  __shared__ float cl[DIM];

  const int cls  = blockIdx.x >> 2;     // 4 blocks per class
  const int tile = blockIdx.x & 3;      // 128 k-rows per block
  const int tid  = threadIdx.x;

  // Stage c (8 KB) into LDS. Trip count uniform -> EXEC stays all-1s for WMMA.
#pragma unroll
  for (int i = tid; i < DIM; i += 256) cl[i] = cvec[cls * DIM + i];
  __syncthreads();

  const int wave = tid >> 5;
  const int lane = tid & 31;
  const int k0   = tile * 128 + wave * 16;
  const int m    = lane & 15;           // k-row within tile (A layout)
  const int kk   = (lane >> 4) * 2;     // d-subcolumn pair: lanes 0-15 -> {0,1}, 16-31 -> {2,3}

  const float* zr = z + ((size_t)cls * KSH + (size_t)(k0 + m)) * DIM;

  v8f acc = {};
  for (int d0 = 0; d0 < DIM; d0 += 4) {
    const int off = d0 + kk;
    v2f A = *(const v2f*)(zr + off);    // A: 16x4 f32 tile of Z
    v2f B = *(const v2f*)(cl + off);    // B: c segment, same column for all N
    acc = __builtin_amdgcn_wmma_f32_16x16x4_f32(
        /*neg_a=*/false, A, /*neg_b=*/false, B,
        /*c_mod=*/(short)0, acc, /*reuse_a=*/false, /*reuse_b=*/false);
  }

  // Column 0 holders: lane 0 has rows k0..k0+7 in acc[0..7], lane 16 rows k0+8..k0+15.
  if (m == 0) {
    const int rbase = cls * KSH + k0 + (lane >> 4) * 8;
#pragma unroll
    for (int r = 0; r < 8; ++r) {
      b[rbase + r] += s[rbase + r] * acc[r];
    }
  }
}

// ---------------------------------------------------------------------------
// Launch: scales ; then [ c-pass ; b-update ] x3 with the final b-update
// skipped (reference discards it). d_out holds c each iteration; the last
// write is the answer. ws layout: s[C*K] then b[C*K] (256 KB total).
// ---------------------------------------------------------------------------
extern "C" void kernel_launch(void* const* d_in, const int* in_sizes, int n_in,
                              void* d_out, int out_size, void* d_ws, size_t ws_size,
                              hipStream_t stream) {
  (void)in_sizes; (void)n_in; (void)out_size; (void)ws_size;

  const float* z = (const float*)d_in[0];
  float* s = (float*)d_ws;
  float* b = s + CLS * KSH;
  float* cout = (float*)d_out;

  scales_kernel<<<(CLS * KSH) / 8, 256, 0, stream>>>(z, s, b);

  for (int it = 0; it < 3; ++it) {
    cvec_kernel<<<CLS, 1024, 0, stream>>>(z, s, b, cout);
    if (it < 2)
      bupd_kernel<<<CLS * 4, 256, 0, stream>>>(z, s, cout, b);
  }
}